// DSS_22720376996350
// MI455X (gfx1250) — compile-verified
//
#include <hip/hip_runtime.h>
#include <hip/hip_bf16.h>

// Problem constants (match reference setup_inputs)
#define B_  8
#define L_  4096
#define H_  512
#define N_  512
#define QQ  64              // chunk length
#define NC_ (L_/QQ)         // 64 chunks
#define BL_ ((size_t)B_*L_) // 32768

typedef __attribute__((ext_vector_type(2))) float v2f;
typedef __attribute__((ext_vector_type(4))) float v4f;
typedef __attribute__((ext_vector_type(8))) float v8f;

// -------- workspace layout (floats), ~131 MB --------
#define OFF_UT   0                               // [H][B*L] transposed u
#define OFF_OT   (OFF_UT  + (size_t)H_*BL_)      // [H][B*L] transposed y
#define OFF_CTR  (OFF_OT  + (size_t)H_*BL_)      // H*N
#define OFF_CTI  (OFF_CTR + (size_t)H_*N_)       // H*N
#define OFF_ER   (OFF_CTI + (size_t)H_*N_)       // N*Q   Er[n][j] = Re(z^(Q-1-j))
#define OFF_EI   (OFF_ER  + (size_t)N_*QQ)       // N*Q
#define OFF_ZR   (OFF_EI  + (size_t)N_*QQ)       // Q*N   Zr[i][n] = Re(z^(i+1))
#define OFF_ZIN  (OFF_ZR  + (size_t)QQ*N_)       // Q*N   Zin[i][n] = -Im(z^(i+1))
#define OFF_ZQR  (OFF_ZIN + (size_t)QQ*N_)       // N     z^Q
#define OFF_ZQI  (OFF_ZQR + (size_t)N_)          // N
#define OFF_KQ   (OFF_ZQI + (size_t)N_)          // H*Q   first Q conv taps

// ============ K0a: modal tables ============
__global__ __launch_bounds__(256) void dss_tables(
    const float* __restrict__ Lr, const float* __restrict__ Li,
    const float* __restrict__ Creal, const float* __restrict__ Cimag,
    float* __restrict__ Ctr, float* __restrict__ Cti,
    float* __restrict__ Er, float* __restrict__ Ei,
    float* __restrict__ Zr, float* __restrict__ Zin,
    float* __restrict__ zQr, float* __restrict__ zQi) {
  int idx = blockIdx.x * 256 + threadIdx.x;     // grid covers H_*N_
  if (idx < H_ * N_) {
    int n = idx % N_;
    float lr = -__expf(Lr[n]), li = __expf(Li[n]);
    float el = __expf(lr);
    float er = el * __cosf(li) - 1.0f;          // exp(lam)-1
    float ei = el * __sinf(li);
    float den = lr * lr + li * li;
    float qr = (er * lr + ei * li) / den;       // (exp(lam)-1)/lam
    float qi = (ei * lr - er * li) / den;
    float cr = Creal[idx], ci = Cimag[idx];
    Ctr[idx] = cr * qr - ci * qi;
    Cti[idx] = cr * qi + ci * qr;
  }
  if (idx < N_ * QQ) {
    { int n = idx / QQ, j = idx % QQ;
      float lr = -__expf(Lr[n]), li = __expf(Li[n]);
      float p = (float)(QQ - 1 - j);
      float mag = __expf(lr * p);
      Er[idx] = mag * __cosf(li * p);
      Ei[idx] = mag * __sinf(li * p); }
    { int n = idx % N_;
      float lr = -__expf(Lr[n]), li = __expf(Li[n]);
      float p = (float)(idx / N_ + 1);
      float mag = __expf(lr * p);
      Zr[idx]  =  mag * __cosf(li * p);
      Zin[idx] = -mag * __sinf(li * p); }
  }
  if (idx < N_) {
    float lr = -__expf(Lr[idx]), li = __expf(Li[idx]);
    float mag = __expf(lr * (float)QQ);
    zQr[idx] = mag * __cosf(li * (float)QQ);
    zQi[idx] = mag * __sinf(li * (float)QQ);
  }
}

// ============ K0b: first Q conv taps  K[h][t] = Re(sum_n Ct z^t) ============
__global__ __launch_bounds__(256) void dss_kq(
    const float* __restrict__ Ctr, const float* __restrict__ Cti,
    const float* __restrict__ Er, const float* __restrict__ Ei,
    float* __restrict__ Kq) {
  int idx = blockIdx.x * 256 + threadIdx.x;     // H_*QQ
  int h = idx / QQ, t = idx % QQ;
  int jj = QQ - 1 - t;                          // Er[n][jj] = Re(z^t)
  float acc = 0.f;
  for (int n = 0; n < N_; ++n)
    acc += Ctr[h * N_ + n] * Er[n * QQ + jj] - Cti[h * N_ + n] * Ei[n * QQ + jj];
  Kq[idx] = acc;
}

// ============ K1: LayerNorm + transpose to ut[H][B*L] ============
__global__ __launch_bounds__(256) void dss_ln(
    const float* __restrict__ x, const float* __restrict__ w,
    const float* __restrict__ bias, float* __restrict__ ut) {
  __shared__ float tile[32 * 513];
  int bI = blockIdx.x;
  int b  = bI >> 7;
  int l0 = (bI & 127) * 32;
  int wv = threadIdx.x >> 5, lane = threadIdx.x & 31;
  for (int rr = 0; rr < 4; ++rr) {
    int row = wv * 4 + rr;                           // 0..31
    const float* xp = x + ((size_t)b * L_ + l0 + row) * H_;
    float vals[16];
    float s = 0.f;
    for (int k = 0; k < 16; ++k) { vals[k] = xp[lane + k * 32]; s += vals[k]; }
    for (int off = 16; off; off >>= 1) s += __shfl_xor(s, off, 32);
    float mean = s * (1.0f / H_);
    float vs = 0.f;
    for (int k = 0; k < 16; ++k) { float d = vals[k] - mean; vs += d * d; }
    for (int off = 16; off; off >>= 1) vs += __shfl_xor(vs, off, 32);
    float inv = rsqrtf(vs * (1.0f / H_) + 1e-5f);
    for (int k = 0; k < 16; ++k) {
      int hc = lane + k * 32;
      tile[row * 513 + hc] = (vals[k] - mean) * inv * w[hc] + bias[hc];
    }
  }
  __syncthreads();
  for (int e = threadIdx.x; e < 32 * H_; e += 256) {
    int h = e >> 5, ll = e & 31;                     // 32 l per h row, coalesced
    ut[(size_t)h * BL_ + (size_t)b * L_ + l0 + ll] = tile[ll * 513 + h];
  }
}

// ============ K2: intra-chunk causal Toeplitz conv via f32 WMMA ============
// A-fragments from REVERSED zero-padded taps so each (a0,a1) pair is an
// ascending contiguous LDS pair (no swap movs); B from bank-safe LDS tile.
__global__ __launch_bounds__(256) void dss_intra(
    const float* __restrict__ ut, const float* __restrict__ Kq,
    float* __restrict__ ot) {
  __shared__ float Krev[128];          // [0..63] = K[h][63-d], [64..127] = 0
  __shared__ float Utile[128 * 68];    // [col][j], stride 68: bank-safe, 16B align
  const int tid = threadIdx.x;
  const int w = tid >> 5, lane = tid & 31;
  const int halfL = lane >> 4, m = lane & 15;
  const int bI = blockIdx.x;           // 8192 = h(512) * it(4) * ctgrp(4)
  const int colB = (bI & 3) * 128;     // 128-col group
  const int it = (bI >> 2) & 3;
  const int h  = bI >> 4;

  if (tid < 128) Krev[tid] = (tid < 64) ? Kq[h * QQ + 63 - tid] : 0.f;
  for (int e = tid; e < 2048; e += 256) {            // 2048 float4 = 8192 floats
    int col_l = e >> 4;
    int j = (e & 15) * 4;
    int col = colB + col_l;
    int bidx = col >> 6, cc = col & 63;
    const float* src = ut + (size_t)h * BL_ + (size_t)bidx * L_ + cc * QQ + j;
    *(v4f*)&Utile[col_l * 68 + j] = *(const v4f*)src;
  }
  __syncthreads();

  const int colIdx = w * 16 + m;       // this lane's local column
  const int irow = it * 16 + m;        // A row (output index i)
  v8f acc = {};
  const int nsteps = (it + 1) * 4;     // triangular: j tiles only up to i
  for (int s = 0; s < nsteps; ++s) {
    int jA = s * 4 + 2 * halfL;
    int dd = 63 - irow + jA;           // ascending pair: a0 @ dd, a1 @ dd+1
    v2f a, bv;
    a[0] = Krev[dd];                   // T[i][jA]   = K[h][irow-jA] or 0
    a[1] = Krev[dd + 1];               // T[i][jA+1]
    bv = *(const v2f*)&Utile[colIdx * 68 + jA];
    acc = __builtin_amdgcn_wmma_f32_16x16x4_f32(false, a, false, bv,
                                                (short)0, acc, false, false);
  }
  int col = colB + colIdx;
  int bidx = col >> 6, cc = col & 63;
  size_t obase = (size_t)h * BL_ + (size_t)bidx * L_ + cc * QQ + it * 16 + 8 * halfL;
  v4f lo = {acc[0], acc[1], acc[2], acc[3]};
  v4f hi = {acc[4], acc[5], acc[6], acc[7]};
  *(v4f*)(ot + obase)     = lo;        // 8 consecutive i per lane, 2x b128
  *(v4f*)(ot + obase + 4) = hi;
}

// ============ K3: inter-chunk scan; state S' = Ct.*S resident in LDS ============
__global__ __launch_bounds__(256) void dss_scan(
    const float* __restrict__ ut,
    const float* __restrict__ Er, const float* __restrict__ Ei,
    const float* __restrict__ Zr, const float* __restrict__ Zin,
    const float* __restrict__ zQr, const float* __restrict__ zQi,
    const float* __restrict__ Ctr, const float* __restrict__ Cti,
    const float* __restrict__ Dv, float* __restrict__ ot) {
  __shared__ float Sr[16 * 516];    // [hh][n] pad 516: bank-safe, v4f aligned
  __shared__ float Si[16 * 516];
  __shared__ float Uc[16 * 68];     // [hh][j]
  __shared__ float Yt[16 * 68];     // [hh][i]
  const int tid = threadIdx.x;
  const int w = tid >> 5;
  const int lane = tid & 31;
  const int halfL = lane >> 4;
  const int m = lane & 15;
  const int b  = blockIdx.x >> 5;
  const int h0 = (blockIdx.x & 31) * 16;
  for (int e = tid; e < 16 * 516; e += 256) { Sr[e] = 0.f; Si[e] = 0.f; }
  __syncthreads();

  const int itile = w & 3;          // wave's 16-row i tile
  const int chalf = w >> 2;         // wave's half of the N contraction

  for (int c = 0; c < NC_; ++c) {
    // ---- A: stage u chunk (coalesced 256B runs from ut) ----
    for (int e = tid; e < QQ * 16; e += 256) {
      int hh = e >> 6, j = e & 63;
      Uc[hh * 68 + j] = ut[(size_t)(h0 + hh) * BL_ + (size_t)b * L_ + c * QQ + j];
    }
    __syncthreads();
    // ---- B: dS tiles (4 n-tiles per wave) into registers ----
    v8f dR[4], dI[4];
    for (int t4 = 0; t4 < 4; ++t4) {
      int n0 = (w + t4 * 8) * 16;
      v8f aR = {}, aI = {};
      for (int s = 0; s < QQ / 4; ++s) {
        int j0 = s * 4 + 2 * halfL;
        v2f ar = *(const v2f*)&Er[(size_t)(n0 + m) * QQ + j0];
        v2f ai = *(const v2f*)&Ei[(size_t)(n0 + m) * QQ + j0];
        v2f bv = *(const v2f*)&Uc[m * 68 + j0];
        aR = __builtin_amdgcn_wmma_f32_16x16x4_f32(false, ar, false, bv,
                                                   (short)0, aR, false, false);
        aI = __builtin_amdgcn_wmma_f32_16x16x4_f32(false, ai, false, bv,
                                                   (short)0, aI, false, false);
      }
      dR[t4] = aR; dI[t4] = aI;
    }
    // ---- C: Y = Zr@S'r + Zin@S'i from OLD state ----
    v8f yacc = {};
    int i0 = itile * 16;
    for (int s = 0; s < 64; ++s) {
      int k0 = chalf * 256 + s * 4 + 2 * halfL;
      v2f a  = *(const v2f*)&Zr[(size_t)(i0 + m) * N_ + k0];
      v2f bv = *(const v2f*)&Sr[m * 516 + k0];
      yacc = __builtin_amdgcn_wmma_f32_16x16x4_f32(false, a, false, bv,
                                                   (short)0, yacc, false, false);
      a  = *(const v2f*)&Zin[(size_t)(i0 + m) * N_ + k0];
      bv = *(const v2f*)&Si[m * 516 + k0];
      yacc = __builtin_amdgcn_wmma_f32_16x16x4_f32(false, a, false, bv,
                                                   (short)0, yacc, false, false);
    }
    if (chalf == 0)
      for (int r = 0; r < 8; ++r) Yt[m * 68 + i0 + r + 8 * halfL] = yacc[r];
    __syncthreads();
    if (chalf == 1)
      for (int r = 0; r < 8; ++r) Yt[m * 68 + i0 + r + 8 * halfL] += yacc[r];
    __syncthreads();                     // all S' reads done, Yt complete
    // ---- D: S' = z^Q * S' + Ct .* dS  (complex, vectorized b128) ----
    for (int t4 = 0; t4 < 4; ++t4) {
      int nb = (w + t4 * 8) * 16 + 8 * halfL;   // 8 consecutive n per lane
      int a2 = m * 516 + nb;
      size_t cb = (size_t)(h0 + m) * N_ + nb;
      v4f srl = *(v4f*)&Sr[a2],        srh = *(v4f*)&Sr[a2 + 4];
      v4f sil = *(v4f*)&Si[a2],        sih = *(v4f*)&Si[a2 + 4];
      v4f zrl = *(const v4f*)&zQr[nb], zrh = *(const v4f*)&zQr[nb + 4];
      v4f zil = *(const v4f*)&zQi[nb], zih = *(const v4f*)&zQi[nb + 4];
      v4f crl = *(const v4f*)&Ctr[cb], crh = *(const v4f*)&Ctr[cb + 4];
      v4f cil = *(const v4f*)&Cti[cb], cih = *(const v4f*)&Cti[cb + 4];
      v4f nsrl, nsil, nsrh, nsih;
      for (int r = 0; r < 4; ++r) {
        float dRl = dR[t4][r], dIl = dI[t4][r];
        float dRh = dR[t4][r + 4], dIh = dI[t4][r + 4];
        nsrl[r] = zrl[r] * srl[r] - zil[r] * sil[r] + (crl[r] * dRl - cil[r] * dIl);
        nsil[r] = zrl[r] * sil[r] + zil[r] * srl[r] + (crl[r] * dIl + cil[r] * dRl);
        nsrh[r] = zrh[r] * srh[r] - zih[r] * sih[r] + (crh[r] * dRh - cih[r] * dIh);
        nsih[r] = zrh[r] * sih[r] + zih[r] * srh[r] + (crh[r] * dIh + cih[r] * dRh);
      }
      *(v4f*)&Sr[a2]     = nsrl;  *(v4f*)&Sr[a2 + 4] = nsrh;
      *(v4f*)&Si[a2]     = nsil;  *(v4f*)&Si[a2 + 4] = nsih;
    }
    // ---- E: ot += Y + D .* u (coalesced along l) ----
    for (int e = tid; e < QQ * 16; e += 256) {
      int hh = e >> 6, i = e & 63;
      size_t g = (size_t)(h0 + hh) * BL_ + (size_t)b * L_ + c * QQ + i;
      ot[g] += Yt[hh * 68 + i] + Dv[h0 + hh] * Uc[hh * 68 + i];
    }
    __syncthreads();                     // protect Uc/S' before next chunk
  }
}

// ============ K4: transpose ot[H][B*L] -> d_out[B*L][H] ============
__global__ __launch_bounds__(256) void dss_tr(
    const float* __restrict__ ot, float* __restrict__ out) {
  __shared__ float t[32][33];
  int h0 = (blockIdx.x & 15) * 32;         // H/32 = 16
  int l0 = (blockIdx.x >> 4) * 32;         // B*L/32 = 1024
  int tl = threadIdx.x & 31, tw = threadIdx.x >> 5;
  for (int r = tw; r < 32; r += 8)
    t[r][tl] = ot[(size_t)(h0 + r) * BL_ + l0 + tl];
  __syncthreads();
  for (int r = tw; r < 32; r += 8)
    out[(size_t)(l0 + r) * H_ + h0 + tl] = t[tl][r];
}

extern "C" void kernel_launch(void* const* d_in, const int* in_sizes, int n_in,
                              void* d_out, int out_size, void* d_ws, size_t ws_size,
                              hipStream_t stream) {
  const float* x  = (const float*)d_in[0];
  const float* lw = (const float*)d_in[1];
  const float* lb = (const float*)d_in[2];
  const float* Lr = (const float*)d_in[3];
  const float* Li = (const float*)d_in[4];
  const float* Cr = (const float*)d_in[5];
  const float* Ci = (const float*)d_in[6];
  const float* Dv = (const float*)d_in[7];
  float* out = (float*)d_out;
  float* ws  = (float*)d_ws;

  float* ut   = ws + OFF_UT;
  float* ot   = ws + OFF_OT;
  float* Ctr  = ws + OFF_CTR;
  float* Cti  = ws + OFF_CTI;
  float* Er   = ws + OFF_ER;
  float* Ei   = ws + OFF_EI;
  float* Zr   = ws + OFF_ZR;
  float* Zin  = ws + OFF_ZIN;
  float* zQr  = ws + OFF_ZQR;
  float* zQi  = ws + OFF_ZQI;
  float* Kq   = ws + OFF_KQ;

  dss_tables<<<(H_ * N_) / 256, 256, 0, stream>>>(Lr, Li, Cr, Ci, Ctr, Cti,
                                                  Er, Ei, Zr, Zin, zQr, zQi);
  dss_kq<<<(H_ * QQ) / 256, 256, 0, stream>>>(Ctr, Cti, Er, Ei, Kq);
  dss_ln<<<B_ * (L_ / 32), 256, 0, stream>>>(x, lw, lb, ut);
  dss_intra<<<H_ * 4 * 4, 256, 0, stream>>>(ut, Kq, ot);
  dss_scan<<<B_ * (H_ / 16), 256, 0, stream>>>(ut, Er, Ei, Zr, Zin, zQr, zQi,
                                               Ctr, Cti, Dv, ot);
  dss_tr<<<16 * (int)(BL_ / 32), 256, 0, stream>>>(ot, out);
}